// GMABSE3_59407987638998
// MI455X (gfx1250) — compile-verified
//
#include <hip/hip_runtime.h>

// Problem constants (match reference)
#define NNODE 10000
#define NEDGE 320000
#define MCH   32
#define NH    8
// per-head feature dim = 16 (4 scalar + 4*3 vector), scale = 1/sqrt(128)

typedef float v2f __attribute__((ext_vector_type(2)));
typedef float v8f __attribute__((ext_vector_type(8)));

// Order-preserving float -> unsigned key (for atomicMax-based segment max).
__device__ __forceinline__ unsigned f2key(float x) {
    unsigned b = __float_as_uint(x);
    return (b & 0x80000000u) ? ~b : (b | 0x80000000u);
}
__device__ __forceinline__ float key2f(unsigned k) {
    if (k == 0u) return 0.0f;  // untouched slot == empty segment -> reference maps to 0
    unsigned b = (k & 0x80000000u) ? (k & 0x7fffffffu) : ~k;
    return __uint_as_float(b);
}

__global__ void gmab_init(float* __restrict__ out, int out_n,
                          unsigned* __restrict__ emaxKey,
                          float* __restrict__ denom, int nh) {
    int i = blockIdx.x * blockDim.x + threadIdx.x;
    if (i < out_n) out[i] = 0.0f;
    if (i < nh) { emaxKey[i] = 0u; denom[i] = 0.0f; }
}

// Pass 1: per-wave tile of 16 edges. For each head, 4 x V_WMMA_F32_16X16X4_F32
// accumulate K_tile(16x4k) * Q_tile(4kx16); diagonal of D = per-edge dot.
__global__ __launch_bounds__(256)
void gmab_dot_max(const float* __restrict__ k0, const float* __restrict__ k1,
                  const float* __restrict__ q0, const float* __restrict__ q1,
                  const int*   __restrict__ dst,
                  float* __restrict__ eArr, unsigned* __restrict__ emaxKey) {
    const int  lane  = threadIdx.x & 31;
    const int  wave  = threadIdx.x >> 5;
    const long ebase = ((long)blockIdx.x * 8 + wave) * 16;
    const int  r     = lane & 15;           // edge-in-tile for A row / B col
    const long e     = ebase + r;
    const int  n     = dst[e];
    const int  klo   = (lane < 16) ? 0 : 2; // K-pair select per ISA f32 A/B layout
    const float scale = 0.08838834764831845f; // 1/sqrt(128)

    #pragma unroll
    for (int h = 0; h < NH; ++h) {
        v8f acc = {};
        #pragma unroll
        for (int kk = 0; kk < 4; ++kk) {
            const int f0 = kk * 4 + klo;    // feature index; kk==0 -> k0/q0, else k1/q1
            v2f a, b;
            if (kk == 0) {
                a.x = k0[e * MCH + h * 4 + f0];
                a.y = k0[e * MCH + h * 4 + f0 + 1];
                b.x = q0[(long)n * MCH + h * 4 + f0];
                b.y = q0[(long)n * MCH + h * 4 + f0 + 1];
            } else {
                a.x = k1[e * (MCH * 3) + h * 12 + (f0 - 4)];
                a.y = k1[e * (MCH * 3) + h * 12 + (f0 - 3)];
                b.x = q1[(long)n * (MCH * 3) + h * 12 + (f0 - 4)];
                b.y = q1[(long)n * (MCH * 3) + h * 12 + (f0 - 3)];
            }
            acc = __builtin_amdgcn_wmma_f32_16x16x4_f32(
                false, a, false, b, (short)0, acc, false, false);
        }
        // Diagonal extraction: D[m][m] at lane m (VGPR m) for m<8,
        // and at lane m+16 (VGPR m-8) for m>=8.
        float ev = 0.0f;
        #pragma unroll
        for (int v = 0; v < 8; ++v) {
            if (lane == v || lane == v + 24) ev = acc[v];
        }
        if (lane < 8 || lane >= 24) {
            const long eo = ebase + ((lane < 8) ? lane : (lane - 16));
            const float es = ev * scale;
            eArr[eo * NH + h] = es;
            atomicMax(&emaxKey[(long)n * NH + h], f2key(es));
        }
    }
}

// Pass 2: ex = exp(e - emax[dst]) in-place; denom += ex.
__global__ void gmab_exp_sum(const int* __restrict__ dst,
                             const unsigned* __restrict__ emaxKey,
                             float* __restrict__ eArr,
                             float* __restrict__ denom) {
    long i = (long)blockIdx.x * blockDim.x + threadIdx.x;
    if (i >= (long)NEDGE * NH) return;
    long e = i / NH;
    int  h = (int)(i % NH);
    int  n = dst[e];
    float em = key2f(emaxKey[(long)n * NH + h]);
    float ex = __expf(eArr[i] - em);
    eArr[i] = ex;
    atomicAdd(&denom[(long)n * NH + h], ex);
}

// Pass 3: one wave per edge (lane = channel m). a = ex/denom; scatter a*v.
__global__ void gmab_scatter(const float* __restrict__ v0, const float* __restrict__ v1,
                             const int*   __restrict__ dst,
                             const float* __restrict__ eArr,
                             const float* __restrict__ denom,
                             float* __restrict__ out) {
    long i = (long)blockIdx.x * blockDim.x + threadIdx.x;
    if (i >= (long)NEDGE * MCH) return;
    long e = i >> 5;
    int  m = (int)(i & 31);
    int  h = m >> 2;
    int  n = dst[e];
    float a = eArr[e * NH + h] / denom[(long)n * NH + h];

    atomicAdd(&out[(long)n * MCH + m], a * v0[i]);

    float* out1 = out + (long)NNODE * MCH;
    long vi = i * 3;
    long oi = ((long)n * MCH + m) * 3;
    atomicAdd(&out1[oi + 0], a * v1[vi + 0]);
    atomicAdd(&out1[oi + 1], a * v1[vi + 1]);
    atomicAdd(&out1[oi + 2], a * v1[vi + 2]);
}

extern "C" void kernel_launch(void* const* d_in, const int* in_sizes, int n_in,
                              void* d_out, int out_size, void* d_ws, size_t ws_size,
                              hipStream_t stream) {
    const float* v0 = (const float*)d_in[0];
    const float* v1 = (const float*)d_in[1];
    const float* k0 = (const float*)d_in[2];
    const float* k1 = (const float*)d_in[3];
    const float* q0 = (const float*)d_in[4];
    const float* q1 = (const float*)d_in[5];
    const int*   dst = (const int*)d_in[6];
    float* out = (float*)d_out;

    // Workspace layout: eArr[E*H] | emaxKey[N*H] | denom[N*H]  (~10.9 MB)
    float*    eArr    = (float*)d_ws;
    unsigned* emaxKey = (unsigned*)(eArr + (size_t)NEDGE * NH);
    float*    denom   = (float*)(emaxKey + (size_t)NNODE * NH);

    const int nh = NNODE * NH;
    const int initN = (out_size > nh) ? out_size : nh;
    gmab_init<<<(initN + 255) / 256, 256, 0, stream>>>(out, out_size, emaxKey, denom, nh);

    gmab_dot_max<<<NEDGE / 128, 256, 0, stream>>>(k0, k1, q0, q1, dst, eArr, emaxKey);

    const long p2 = (long)NEDGE * NH;
    gmab_exp_sum<<<(int)((p2 + 255) / 256), 256, 0, stream>>>(dst, emaxKey, eArr, denom);

    const long p3 = (long)NEDGE * MCH;
    gmab_scatter<<<(int)((p3 + 255) / 256), 256, 0, stream>>>(v0, v1, dst, eArr, denom, out);
}